// SGMConfidenceTokenRouter_19945828122937
// MI455X (gfx1250) — compile-verified
//
#include <hip/hip_runtime.h>
#include <hip/hip_bf16.h>
#include <cstdint>

#define GRID   14
#define NTOK   196          // GRID*GRID
#define HIMG   224
#define WIMG   224
#define PATCH  16
#define THRESH 0.6f

typedef float vfloat4 __attribute__((ext_vector_type(4)));  // native clang vec

// ---------------------------------------------------------------------------
// Kernel B (placed first for disasm visibility): token gather.
// One wave (32 lanes) per output row (b, p).
//   valid rows: stream 768 f32 (192 x B128 packets) through LDS using the
//               CDNA5 async global<->LDS DMA path (ASYNCcnt-tracked).
//   padded rows: zero-fill with non-temporal B128 stores.
// Validity is wave-uniform, so EXEC is all-ones around the async ops.
// ---------------------------------------------------------------------------
__global__ __launch_bounds__(256)
void gather_tokens_kernel(const float* __restrict__ tokens,
                          const float* __restrict__ order_f,  // B*196
                          const float* __restrict__ keep_f,   // B
                          float* __restrict__ out,            // B*196*D
                          int nrows, int D) {
  __shared__ __align__(16) float lbuf[8 * 768];   // 3 KB per wave, 24 KB/block

  const int lane = threadIdx.x & 31;
  const int wid  = threadIdx.x >> 5;
  const int r    = blockIdx.x * 8 + wid;
  if (r >= nrows) return;

  const int b   = r / NTOK;
  const int p   = r - b * NTOK;
  const int src = (int)order_f[r];      // exact small int stored as float
  const int kc  = (int)keep_f[b];
  const int nq  = D >> 2;               // float4 packets per row (192)

  vfloat4* dst = (vfloat4*)(out + (size_t)r * (size_t)D);

  if (p < kc) {
    const vfloat4* srcp =
        (const vfloat4*)(tokens + ((size_t)b * NTOK + src) * (size_t)D);
    // LDS byte offset of this wave's staging buffer (low 32 bits of flat addr).
    const unsigned ldsBase = (unsigned)(uintptr_t)(&lbuf[wid * 768]);

    for (int base = 0; base < nq; base += 192) {
      int chunk = nq - base; if (chunk > 192) chunk = 192;
      // Phase 1: async DMA global -> LDS, 16B per lane-op, ASYNCcnt-tracked.
      for (int k = lane; k < chunk; k += 32) {
        unsigned lds_addr = ldsBase + (unsigned)k * 16u;
        uint64_t gaddr    = (uint64_t)(uintptr_t)(srcp + base + k);
        asm volatile("global_load_async_to_lds_b128 %0, %1, off"
                     :: "v"(lds_addr), "v"(gaddr) : "memory");
      }
      asm volatile("s_wait_asynccnt 0" ::: "memory");
      // Phase 2: async DMA LDS -> global.
      for (int k = lane; k < chunk; k += 32) {
        unsigned lds_addr = ldsBase + (unsigned)k * 16u;
        uint64_t gaddr    = (uint64_t)(uintptr_t)(dst + base + k);
        asm volatile("global_store_async_from_lds_b128 %0, %1, off"
                     :: "v"(gaddr), "v"(lds_addr) : "memory");
      }
      asm volatile("s_wait_asynccnt 0" ::: "memory");  // LDS reuse safety
    }
  } else {
    const vfloat4 z = {0.f, 0.f, 0.f, 0.f};
    for (int k = lane; k < nq; k += 32)
      __builtin_nontemporal_store(z, dst + k);  // write-once padding, skip L2 RT
  }
}

// ---------------------------------------------------------------------------
// Kernel A: clip + 16x16 avg-pool -> conf_grid, stable partition -> order,
// keep_counts, and global high-token count (integer atomic: deterministic).
// One block per image; 196 of 256 threads each own one pooled cell.
// ---------------------------------------------------------------------------
__global__ __launch_bounds__(256)
void pool_mask_order_kernel(const float* __restrict__ conf,
                            float* __restrict__ conf_grid_out,  // B*196
                            float* __restrict__ order_out,      // B*196 (ints as f32)
                            float* __restrict__ keep_out,       // B     (ints as f32)
                            int*   __restrict__ high_total) {
  __shared__ int smask[NTOK];
  __shared__ int sorder[NTOK];

  const int b = blockIdx.x;
  const int t = threadIdx.x;
  const float* base = conf + (size_t)b * (HIMG * WIMG);

  if (t < NTOK) {
    const int cy = t / GRID, cx = t % GRID;
    const float* p0 = base + (size_t)(cy * PATCH) * WIMG + cx * PATCH;
    float sum = 0.f;
    for (int i = 0; i < PATCH; ++i) {
      const vfloat4* row = (const vfloat4*)(p0 + (size_t)i * WIMG);  // 16B aligned
#pragma unroll
      for (int j = 0; j < 4; ++j) {
        vfloat4 v = row[j];
        sum += fminf(fmaxf(v.x, 0.f), 1.f) + fminf(fmaxf(v.y, 0.f), 1.f) +
               fminf(fmaxf(v.z, 0.f), 1.f) + fminf(fmaxf(v.w, 0.f), 1.f);
      }
    }
    const float mean = sum * (1.0f / (PATCH * PATCH));
    conf_grid_out[b * NTOK + t] = mean;
    smask[t] = (mean > THRESH) ? 1 : 0;
  }
  __syncthreads();

  if (t == 0) {
    // Stable partition == jnp.argsort(bool, stable): kept (0) first, pruned (1) after.
    int kc = 0;
    for (int n = 0; n < NTOK; ++n) if (!smask[n]) sorder[kc++] = n;
    int pos = kc;
    for (int n = 0; n < NTOK; ++n) if (smask[n]) sorder[pos++] = n;
    keep_out[b] = (float)kc;
    atomicAdd(high_total, NTOK - kc);   // integer atomic: order-independent
  }
  __syncthreads();

  if (t < NTOK) order_out[b * NTOK + t] = (float)sorder[t];
}

// ---------------------------------------------------------------------------
// Kernel C: prune_ratio scalar.
// ---------------------------------------------------------------------------
__global__ void prune_ratio_kernel(const int* __restrict__ high_total,
                                   float* __restrict__ out, float inv_total) {
  out[0] = (float)(*high_total) * inv_total;
}

// ---------------------------------------------------------------------------
// d_out layout (float*, outputs concatenated in reference return order):
//   [0]                keep_tokens  B*196*D
//   [+B*196*D]         conf_grid    B*196
//   [+B*196]           order        B*196   (ints as f32)
//   [+B*196]           keep_counts  B       (ints as f32)
//   [+B]               prune_ratio  1
// ---------------------------------------------------------------------------
extern "C" void kernel_launch(void* const* d_in, const int* in_sizes, int n_in,
                              void* d_out, int out_size, void* d_ws, size_t ws_size,
                              hipStream_t stream) {
  const float* conf   = (const float*)d_in[0];
  const float* tokens = (const float*)d_in[1];

  const int B = in_sizes[0] / (HIMG * WIMG);         // 256
  const int D = in_sizes[1] / (B * NTOK);            // 768

  float* out = (float*)d_out;
  const size_t off_conf  = (size_t)B * NTOK * (size_t)D;
  const size_t off_order = off_conf  + (size_t)B * NTOK;
  const size_t off_keep  = off_order + (size_t)B * NTOK;
  const size_t off_pr    = off_keep  + (size_t)B;

  int* high_total = (int*)d_ws;
  (void)hipMemsetAsync(d_ws, 0, sizeof(int), stream);  // graph-capture safe

  pool_mask_order_kernel<<<B, 256, 0, stream>>>(
      conf, out + off_conf, out + off_order, out + off_keep, high_total);

  const int nrows = B * NTOK;                        // 50176 rows
  gather_tokens_kernel<<<(nrows + 7) / 8, 256, 0, stream>>>(
      tokens, out + off_order, out + off_keep, out, nrows, D);

  prune_ratio_kernel<<<1, 1, 0, stream>>>(
      high_total, out + off_pr, 1.0f / (float)nrows);
}